// SMPL_72739566125552
// MI455X (gfx1250) — compile-verified
//
#include <hip/hip_runtime.h>

// ---------------- problem constants ----------------
#define B_    512
#define V_    6890
#define JNT_  24
#define NB_   10
#define PF_   207
#define NJ_   19
#define V3_   (V_*3)          // 20670
#define KP_   224             // padded K: cols [0..9]=beta/shapedirs, [10..11]=0,
                              //           [12..218]=pose/posedirs,   [219..223]=0
#define BT_   32              // batches per workgroup (heavy kernel)
#define VT_   64              // vertices per workgroup
#define ALDS_STRIDE 225       // odd stride -> conflict-free A-fragment reads

typedef float v2f __attribute__((ext_vector_type(2)));
typedef float v8f __attribute__((ext_vector_type(8)));

__constant__ int c_parents[JNT_] = {0,0,0,0,1,2,3,4,5,6,7,8,9,9,9,12,13,14,16,17,18,19,20,21};

// D = A(16x4) * B(4x16) + C, fp32 WMMA (wave32).
__device__ __forceinline__ v8f wmma4(v2f a, v2f b, v8f c) {
    return __builtin_amdgcn_wmma_f32_16x16x4_f32(
        /*neg_a=*/false, a, /*neg_b=*/false, b,
        /*c_mod=*/(short)0, c, /*reuse_a=*/false, /*reuse_b=*/false);
}

// ---------------- K zero: clear output (atomically accumulated later) -------
__global__ void k_zero(float* __restrict__ out, int n) {
    int i = blockIdx.x * 256 + threadIdx.x;
    if (i < n) out[i] = 0.0f;
}

// ---------------- K0: Rodrigues + build padded A-matrix ---------------------
// one thread per (b, j).  Rs[b*24+j][9], Amat[b][KP_]
__global__ __launch_bounds__(256) void k_rod(const float* __restrict__ theta,
                                             const float* __restrict__ beta,
                                             float* __restrict__ Rs,
                                             float* __restrict__ Amat) {
    int i = blockIdx.x * 256 + threadIdx.x;
    if (i >= B_ * JNT_) return;
    int b = i / JNT_, j = i % JNT_;
    float ax = theta[b*72 + j*3 + 0];
    float ay = theta[b*72 + j*3 + 1];
    float az = theta[b*72 + j*3 + 2];
    float ex = ax + 1e-8f, ey = ay + 1e-8f, ez = az + 1e-8f;
    float angle = sqrtf(ex*ex + ey*ey + ez*ez);
    float inv   = 1.0f / angle;
    float h     = angle * 0.5f;
    float w     = cosf(h);
    float s     = sinf(h);
    float x = ax*inv*s, y = ay*inv*s, z = az*inv*s;
    float R[9];
    R[0] = w*w + x*x - y*y - z*z; R[1] = 2.f*(x*y - w*z);        R[2] = 2.f*(x*z + w*y);
    R[3] = 2.f*(x*y + w*z);       R[4] = w*w - x*x + y*y - z*z;  R[5] = 2.f*(y*z - w*x);
    R[6] = 2.f*(x*z - w*y);       R[7] = 2.f*(y*z + w*x);        R[8] = w*w - x*x - y*y + z*z;
    #pragma unroll
    for (int e = 0; e < 9; ++e) Rs[i*9 + e] = R[e];
    if (j > 0) {
        #pragma unroll
        for (int e = 0; e < 9; ++e)
            Amat[b*KP_ + 12 + (j-1)*9 + e] = R[e] - ((e==0||e==4||e==8) ? 1.0f : 0.0f);
    } else {
        // shape coefficients + exact-zero padding columns (enables clamped B reads)
        #pragma unroll
        for (int k = 0; k < NB_; ++k) Amat[b*KP_ + k] = beta[b*NB_ + k];
        Amat[b*KP_ + 10] = 0.0f;
        Amat[b*KP_ + 11] = 0.0f;
        #pragma unroll
        for (int c = 219; c < KP_; ++c) Amat[b*KP_ + c] = 0.0f;
    }
}

// ---------------- K1: SJ[k][j][c] = sum_v shapedirs[k][v3c]*Jreg[v][j] ------
// blockIdx.x = k*24 + j ; k==10 -> v_template reduction into Jt.
__global__ __launch_bounds__(256) void k_sj(const float* __restrict__ shapedirs,
                                            const float* __restrict__ vtemp,
                                            const float* __restrict__ Jreg,
                                            float* __restrict__ SJ,
                                            float* __restrict__ Jt) {
    int k = blockIdx.x / JNT_;
    int j = blockIdx.x % JNT_;
    float s0 = 0.f, s1 = 0.f, s2 = 0.f;
    for (int v = threadIdx.x; v < V_; v += 256) {
        float w = Jreg[v*JNT_ + j];
        const float* src = (k < NB_) ? (shapedirs + (size_t)k*V3_ + v*3) : (vtemp + v*3);
        s0 += src[0]*w; s1 += src[1]*w; s2 += src[2]*w;
    }
    __shared__ float red[256][3];
    red[threadIdx.x][0] = s0; red[threadIdx.x][1] = s1; red[threadIdx.x][2] = s2;
    __syncthreads();
    for (int off = 128; off > 0; off >>= 1) {
        if ((int)threadIdx.x < off) {
            red[threadIdx.x][0] += red[threadIdx.x+off][0];
            red[threadIdx.x][1] += red[threadIdx.x+off][1];
            red[threadIdx.x][2] += red[threadIdx.x+off][2];
        }
        __syncthreads();
    }
    if (threadIdx.x == 0) {
        float* dst = (k < NB_) ? (SJ + (k*JNT_ + j)*3) : (Jt + j*3);
        dst[0] = red[0][0]; dst[1] = red[0][1]; dst[2] = red[0][2];
    }
}

// ---------------- K2: J = Jt + beta@SJ ; kinematic chain -> A_rel -----------
// 32 threads/block, one batch per thread; chain state in LDS (46 KB).
__global__ __launch_bounds__(32) void k_rigid(const float* __restrict__ beta,
                                              const float* __restrict__ SJ,
                                              const float* __restrict__ Jt,
                                              const float* __restrict__ Rs,
                                              float* __restrict__ Arel) {
    __shared__ float Rg[32][JNT_][9];
    __shared__ float Tg[32][JNT_][3];
    __shared__ float Jl[32][JNT_][3];
    int t = threadIdx.x;
    int b = blockIdx.x * 32 + t;
    float bb[NB_];
    #pragma unroll
    for (int k = 0; k < NB_; ++k) bb[k] = beta[b*NB_ + k];
    for (int j = 0; j < JNT_; ++j) {
        #pragma unroll
        for (int c = 0; c < 3; ++c) {
            float s = Jt[j*3 + c];
            #pragma unroll
            for (int k = 0; k < NB_; ++k) s += bb[k] * SJ[(k*JNT_ + j)*3 + c];
            Jl[t][j][c] = s;
        }
    }
    for (int j = 0; j < JNT_; ++j) {
        float R[9];
        #pragma unroll
        for (int e = 0; e < 9; ++e) R[e] = Rs[((size_t)b*JNT_ + j)*9 + e];
        if (j == 0) {
            #pragma unroll
            for (int e = 0; e < 9; ++e) Rg[t][0][e] = R[e];
            #pragma unroll
            for (int c = 0; c < 3; ++c) Tg[t][0][c] = Jl[t][0][c];
        } else {
            int p = c_parents[j];
            float tb[3];
            #pragma unroll
            for (int c = 0; c < 3; ++c) tb[c] = Jl[t][j][c] - Jl[t][p][c];
            float Rn[9], tn[3];
            #pragma unroll
            for (int r = 0; r < 3; ++r) {
                #pragma unroll
                for (int c = 0; c < 3; ++c)
                    Rn[r*3+c] = Rg[t][p][r*3+0]*R[0*3+c]
                              + Rg[t][p][r*3+1]*R[1*3+c]
                              + Rg[t][p][r*3+2]*R[2*3+c];
                tn[r] = Rg[t][p][r*3+0]*tb[0] + Rg[t][p][r*3+1]*tb[1]
                      + Rg[t][p][r*3+2]*tb[2] + Tg[t][p][r];
            }
            #pragma unroll
            for (int e = 0; e < 9; ++e) Rg[t][j][e] = Rn[e];
            #pragma unroll
            for (int c = 0; c < 3; ++c) Tg[t][j][c] = tn[c];
        }
        // A_rel: 9 rotation + 3 translation (t_global - Rg @ J_j)
        float jx = Jl[t][j][0], jy = Jl[t][j][1], jz = Jl[t][j][2];
        float* o = Arel + ((size_t)b*JNT_ + j)*12;
        #pragma unroll
        for (int e = 0; e < 9; ++e) o[e] = Rg[t][j][e];
        #pragma unroll
        for (int r = 0; r < 3; ++r)
            o[9+r] = Tg[t][j][r] - (Rg[t][j][r*3+0]*jx + Rg[t][j][r*3+1]*jy + Rg[t][j][r*3+2]*jz);
    }
}

// ---------------- K3: fused blendshape GEMM + skinning + joint regression ---
// grid (ceil(V/64), B/32), 256 threads = 8 waves.
// Wave (m,n): 16 batches x 16 vertices; 3 f32 WMMA accumulators (x,y,z).
// Branch-free hot loop: out-of-range K rows / vertex columns are index-CLAMPED;
// the matching A-matrix columns are exact zeros, so clamped (finite) B values
// contribute exactly 0.0f.
__global__ __launch_bounds__(256) void k_lbs(const float* __restrict__ Amat,
                                             const float* __restrict__ shapedirs,
                                             const float* __restrict__ posedirs,
                                             const float* __restrict__ vtemp,
                                             const float* __restrict__ lbsw,
                                             const float* __restrict__ jointreg,
                                             const float* __restrict__ Arel,
                                             float* __restrict__ out) {
    __shared__ float Alds[BT_][ALDS_STRIDE];      // 28.8 KB
    __shared__ float Wl[VT_][JNT_];               //  6.1 KB
    __shared__ float VTl[VT_][3];
    __shared__ float JRl[VT_][NJ_];               //  4.9 KB
    __shared__ float jpart[BT_][NJ_][3];          //  7.3 KB

    const int tid    = threadIdx.x;
    const int vtBase = blockIdx.x * VT_;
    const int btBase = blockIdx.y * BT_;

    for (int i = tid; i < BT_*KP_; i += 256) {
        int r = i / KP_, c = i % KP_;
        Alds[r][c] = Amat[(btBase + r)*KP_ + c];
    }
    for (int i = tid; i < VT_*JNT_; i += 256) {
        int v = vtBase + i / JNT_;
        Wl[i/JNT_][i%JNT_] = (v < V_) ? lbsw[v*JNT_ + i%JNT_] : 0.0f;
    }
    for (int i = tid; i < VT_*3; i += 256) {
        int v = vtBase + i/3;
        VTl[i/3][i%3] = (v < V_) ? vtemp[v*3 + i%3] : 0.0f;
    }
    for (int i = tid; i < VT_*NJ_; i += 256) {
        int v = vtBase + i / NJ_;
        JRl[i/NJ_][i%NJ_] = (v < V_) ? jointreg[v*NJ_ + i%NJ_] : 0.0f;
    }
    for (int i = tid; i < BT_*NJ_*3; i += 256) ((float*)jpart)[i] = 0.0f;
    __syncthreads();

    const int lane  = tid & 31;
    const int wave  = tid >> 5;
    const int waveM = wave >> 2;       // 0..1  (batch sub-tile)
    const int waveN = wave & 3;        // 0..3  (vertex sub-tile)
    const int m     = lane & 15;
    const int half  = lane >> 4;
    const int kpair = half * 2;        // A frag: lanes<16 -> K{0,1}, lanes>=16 -> K{2,3}
    const int arow  = waveM*16 + m;    // A frag M index (local batch)
    const int vcol  = vtBase + waveN*16 + m;           // B frag N index (vertex)
    const int vclamp = (vcol < V_) ? vcol : (V_ - 1);  // clamped (results discarded)
    const size_t colbase = (size_t)vclamp * 3;

    // prefetch this lane's B strips and the A_rel rows used in skinning
    __builtin_prefetch(posedirs + colbase, 0, 3);
    __builtin_prefetch(shapedirs + colbase, 0, 3);
    __builtin_prefetch(Arel + (size_t)(btBase + waveM*16)*JNT_*12, 0, 3);

    v8f acc[3];
    acc[0] = (v8f){0,0,0,0,0,0,0,0};
    acc[1] = (v8f){0,0,0,0,0,0,0,0};
    acc[2] = (v8f){0,0,0,0,0,0,0,0};

    const size_t koff = (size_t)kpair * V3_;   // per-lane K-row pair offset

    // ---- segment 1: shapedirs (K rows 0..9), Amat cols [0..11] ----
    {
        const float* sb = shapedirs + colbase + koff;
        #pragma unroll
        for (int kb = 0; kb < 8; kb += 4) {    // rows 0..7: no clamping needed
            v2f a;
            a.x = Alds[arow][kb + kpair];
            a.y = Alds[arow][kb + kpair + 1];
            const float* p = sb + (size_t)kb * V3_;
            #pragma unroll
            for (int c = 0; c < 3; ++c) {
                v2f bf; bf.x = p[c]; bf.y = p[V3_ + c];
                acc[c] = wmma4(a, bf, acc[c]);
            }
        }
        // tail kb=8: rows 8..11, clamp rows 10,11 -> 9 (A cols 10,11 == 0)
        v2f a;
        a.x = Alds[arow][8 + kpair];
        a.y = Alds[arow][9 + kpair];
        int k0 = 8 + kpair;
        int r0 = (k0     < NB_) ? k0     : (NB_-1);
        int r1 = (k0 + 1 < NB_) ? k0 + 1 : (NB_-1);
        const float* p0 = shapedirs + colbase + (size_t)r0 * V3_;
        const float* p1 = shapedirs + colbase + (size_t)r1 * V3_;
        #pragma unroll
        for (int c = 0; c < 3; ++c) {
            v2f bf; bf.x = p0[c]; bf.y = p1[c];
            acc[c] = wmma4(a, bf, acc[c]);
        }
    }

    // ---- segment 2: posedirs (K rows 0..206), Amat cols [12..219] ----
    {
        const float* pb = posedirs + colbase + koff;
        #pragma unroll
        for (int kb = 12; kb < 216; kb += 4) { // rows 0..203: no clamping needed
            v2f a;
            a.x = Alds[arow][kb + kpair];
            a.y = Alds[arow][kb + kpair + 1];
            const float* p = pb + (size_t)(kb - 12) * V3_;
            #pragma unroll
            for (int c = 0; c < 3; ++c) {
                v2f bf; bf.x = p[c]; bf.y = p[V3_ + c];
                acc[c] = wmma4(a, bf, acc[c]);
            }
        }
        // tail kb=216: rows 204..207, clamp row 207 -> 206 (A col 219 == 0)
        v2f a;
        a.x = Alds[arow][216 + kpair];
        a.y = Alds[arow][217 + kpair];
        int kr = 204 + kpair;
        int r1 = (kr + 1 < PF_) ? kr + 1 : (PF_-1);
        const float* p0 = posedirs + colbase + (size_t)kr * V3_;
        const float* p1 = posedirs + colbase + (size_t)r1 * V3_;
        #pragma unroll
        for (int c = 0; c < 3; ++c) {
            v2f bf; bf.x = p0[c]; bf.y = p1[c];
            acc[c] = wmma4(a, bf, acc[c]);
        }
    }

    // ---- skinning + joint regression (C frag: VGPR r -> M=r+8*half, N=lane&15) ----
    const int vloc = waveN*16 + m;
    const int v    = vtBase + vloc;
    #pragma unroll
    for (int r = 0; r < 8; ++r) {
        const int bloc = waveM*16 + half*8 + r;
        float px = acc[0][r] + VTl[vloc][0];
        float py = acc[1][r] + VTl[vloc][1];
        float pz = acc[2][r] + VTl[vloc][2];
        float T[12];
        #pragma unroll
        for (int e = 0; e < 12; ++e) T[e] = 0.0f;
        const float* Ab = Arel + ((size_t)(btBase + bloc)*JNT_)*12;
        #pragma unroll 1
        for (int j = 0; j < JNT_; ++j) {
            float w = Wl[vloc][j];
            const float* Ap = Ab + j*12;
            #pragma unroll
            for (int e = 0; e < 12; ++e) T[e] += w * Ap[e];
        }
        float vx = T[0]*px + T[1]*py + T[2]*pz + T[9];
        float vy = T[3]*px + T[4]*py + T[5]*pz + T[10];
        float vz = T[6]*px + T[7]*py + T[8]*pz + T[11];
        if (v < V_) {
            #pragma unroll 1
            for (int j = 0; j < NJ_; ++j) {
                float jw = JRl[vloc][j];
                unsafeAtomicAdd(&jpart[bloc][j][0], vx*jw);
                unsafeAtomicAdd(&jpart[bloc][j][1], vy*jw);
                unsafeAtomicAdd(&jpart[bloc][j][2], vz*jw);
            }
        }
    }
    __syncthreads();
    for (int i = tid; i < BT_*NJ_*3; i += 256) {
        int bl = i / (NJ_*3);
        int rest = i % (NJ_*3);
        unsafeAtomicAdd(&out[(btBase + bl)*NJ_*3 + rest], ((float*)jpart)[i]);
    }
}

// ---------------- launch ----------------------------------------------------
extern "C" void kernel_launch(void* const* d_in, const int* in_sizes, int n_in,
                              void* d_out, int out_size, void* d_ws, size_t ws_size,
                              hipStream_t stream) {
    const float* beta      = (const float*)d_in[0];
    const float* theta     = (const float*)d_in[1];
    const float* vtemp     = (const float*)d_in[2];
    const float* shapedirs = (const float*)d_in[3];
    const float* Jreg      = (const float*)d_in[4];
    const float* posedirs  = (const float*)d_in[5];
    const float* lbsw      = (const float*)d_in[6];
    const float* jointreg  = (const float*)d_in[7];
    float* out = (float*)d_out;

    float* ws = (float*)d_ws;
    float* Amat = ws;                                   // 512*224
    float* Rs   = Amat + (size_t)B_*KP_;                // 512*24*9
    float* SJ   = Rs   + (size_t)B_*JNT_*9;             // 10*24*3
    float* Jt   = SJ   + NB_*JNT_*3;                    // 24*3
    float* Arel = Jt   + JNT_*3;                        // 512*24*12

    k_zero<<<(out_size + 255)/256, 256, 0, stream>>>(out, out_size);
    k_rod <<<(B_*JNT_ + 255)/256, 256, 0, stream>>>(theta, beta, Rs, Amat);
    k_sj  <<<(NB_+1)*JNT_, 256, 0, stream>>>(shapedirs, vtemp, Jreg, SJ, Jt);
    k_rigid<<<B_/32, 32, 0, stream>>>(beta, SJ, Jt, Rs, Arel);
    dim3 grid((V_ + VT_ - 1)/VT_, B_/BT_);
    k_lbs <<<grid, 256, 0, stream>>>(Amat, shapedirs, posedirs, vtemp, lbsw, jointreg, Arel, out);
}